// quantize_conv__14267881357571
// MI455X (gfx1250) — compile-verified
//
#include <hip/hip_runtime.h>
#include <stdint.h>

// Nearest-codebook scalar quantization for MI455X (gfx1250).
//   x: [32,56,56,128] fp32 (12,845,056 elems), codebook: [32] fp32.
//   out[i] = codebook[argmin_k |x[i] - codebook[k]|]  (first index wins ties).
//
// Roofline: 103 MB traffic => ~4.4us floor @ 23.3 TB/s. A 32-entry linear scan
// is ~128 VALU instr/elem (~5x over the floor), so we rank-sort the codebook
// once and do a 5-step LDS binary search + exact 2-candidate fp32 distance
// compare with original-index tie-break (bit-exact vs the reference argmin),
// ~28 instr/elem => roughly balanced with the memory floor.

#define KCB 32

typedef int v2i __attribute__((vector_size(8)));   // <2 x i32>, 8 bytes

// ---------------------------------------------------------------------------
// Kernel 1: single-wave32 rank sort. Lane i holds codebook[i]; its rank is
// the number of (value, index) pairs ordered before it (stable: ties broken
// by original index => ranks are unique). Uniform codebook[k] reads become
// SGPR operands, so this is ~3 VALU ops per k, no scratch, runs once.
// ---------------------------------------------------------------------------
__global__ void sort_codebook_kernel(const float* __restrict__ codebook,
                                     float2* __restrict__ pairs) {
  const int lane = (int)threadIdx.x;            // launched with exactly 32
  const float v = codebook[lane];
  int rank = 0;
#pragma unroll
  for (int k = 0; k < KCB; ++k) {
    const float w = codebook[k];                // lane-uniform -> SGPR
    rank += (w < v || (w == v && k < lane)) ? 1 : 0;
  }
  pairs[rank] = make_float2(v, __int_as_float(lane));
}

// ---------------------------------------------------------------------------
// Per-scalar nearest lookup: branchless binary search over sorted values in
// LDS (5x ds_load_b32 + cmp + cndmask), then exact distance compare of the
// two bracketing candidates. Distances are the same fp32 |x-c| the reference
// computes; ties resolved by smaller ORIGINAL index (matches jnp.argmin).
// ---------------------------------------------------------------------------
__device__ __forceinline__ float nearest_scalar(float xv, const float2* lp) {
  int j = 0;
  if (lp[j + 16].x <= xv) j += 16;
  if (lp[j +  8].x <= xv) j += 8;
  if (lp[j +  4].x <= xv) j += 4;
  if (lp[j +  2].x <= xv) j += 2;
  if (lp[j +  1].x <= xv) j += 1;
  // Now candidates are sorted[j] and sorted[j+1] (clamped).
  float2 p0 = lp[j];
  float2 p1 = lp[(j < KCB - 1) ? (j + 1) : j];
  float d0 = __builtin_fabsf(xv - p0.x);
  float d1 = __builtin_fabsf(xv - p1.x);
  int  i0 = __float_as_int(p0.y);
  int  i1 = __float_as_int(p1.y);
  bool take1 = (d1 < d0) || ((d1 == d0) && (i1 < i0));
  return take1 ? p1.x : p0.x;
}

// ---------------------------------------------------------------------------
// Kernel 2: stream x as float4 (global_load_b128 / global_store_b128),
// codebook pairs staged into LDS via the CDNA5 async global->LDS copy
// (ASYNCcnt + s_wait_asynccnt), falling back to a plain ds copy if absent.
// ---------------------------------------------------------------------------
__global__ void quantize_nearest_kernel(const float* __restrict__ xf,
                                        float2* __restrict__ pairs,
                                        float* __restrict__ of,
                                        int n4, int n) {
  __shared__ float2 lp[KCB];
  const int t = threadIdx.x;

  if (t < KCB) {
#if __has_builtin(__builtin_amdgcn_global_load_async_to_lds_b64)
    // CDNA5 async DMA path: one b64 per lane (lanes 0..31), tracked by ASYNCcnt.
    // Signature (from clang diagnostic): (v2i AS1*, v2i AS3*, Imm offset, Imm cpol).
    __builtin_amdgcn_global_load_async_to_lds_b64(
        (__attribute__((address_space(1))) v2i*)(pairs + t),
        (__attribute__((address_space(3))) v2i*)(lp + t),
        /*imm offset=*/0, /*cpol=*/0);
#if __has_builtin(__builtin_amdgcn_s_wait_asynccnt)
    __builtin_amdgcn_s_wait_asynccnt(0);
#else
    asm volatile("s_wait_asynccnt 0" ::: "memory");
#endif
#else
    lp[t] = pairs[t];
#endif
  }
  __syncthreads();

  const float4* __restrict__ x4 = (const float4*)xf;
  float4* __restrict__ o4 = (float4*)of;
  const int stride = (int)(gridDim.x * blockDim.x);

  for (int i = (int)(blockIdx.x * blockDim.x) + t; i < n4; i += stride) {
    // Prefetch one grid-stride ahead (lowers to global_prefetch_b8 on gfx1250).
    if (i + stride < n4) __builtin_prefetch(&x4[i + stride], 0, 0);
    float4 xv = x4[i];
    float4 q;
    q.x = nearest_scalar(xv.x, lp);
    q.y = nearest_scalar(xv.y, lp);
    q.z = nearest_scalar(xv.z, lp);
    q.w = nearest_scalar(xv.w, lp);
    o4[i] = q;
  }

  // Scalar tail (n not divisible by 4; not hit for this shape but kept exact).
  const int tail = n - n4 * 4;
  if (blockIdx.x == 0 && t < tail) {
    const int idx = n4 * 4 + t;
    of[idx] = nearest_scalar(xf[idx], lp);
  }
}

// ---------------------------------------------------------------------------
// Host-side launch. d_in[0]=x (fp32), d_in[1]=codebook (fp32, 32 elems).
// d_out = fp32, out_size = B*W*H*C. d_ws holds 32 float2 pairs (256 B).
// ---------------------------------------------------------------------------
extern "C" void kernel_launch(void* const* d_in, const int* in_sizes, int n_in,
                              void* d_out, int out_size, void* d_ws, size_t ws_size,
                              hipStream_t stream) {
  (void)in_sizes; (void)n_in; (void)ws_size;
  const float* x  = (const float*)d_in[0];
  const float* cb = (const float*)d_in[1];
  float* out = (float*)d_out;
  float2* pairs = (float2*)d_ws;

  sort_codebook_kernel<<<1, KCB, 0, stream>>>(cb, pairs);

  const int n  = out_size;
  const int n4 = n / 4;
  const int threads = 256;                    // 8 wave32 waves per block
  int blocks = (n4 + threads - 1) / threads;  // 12,544 for this shape
  if (blocks > 8192) blocks = 8192;           // grid-stride amortizes LDS fill
  if (blocks < 1) blocks = 1;
  quantize_nearest_kernel<<<blocks, threads, 0, stream>>>(x, pairs, out, n4, n);
}